// WeightAndSum_88708254531843
// MI455X (gfx1250) — compile-verified
//
#include <hip/hip_runtime.h>

// MI455X reasoning: memory-bound (1.02 GB feats stream, ~45 us floor at
// 23.3 TB/s). Gate GEMV via V_WMMA_F32_16X16X4_F32 per 16-row tile; sorted
// segment ids -> register running sum, atomic flush at boundaries only.
// Staging now uses CDNA5 GLOBAL_LOAD_ASYNC_TO_LDS_B128 (ASYNCcnt-tracked DMA,
// no VGPR round trip) with per-wave double buffering so global-load latency
// overlaps WMMA/accumulate compute instead of stalling on s_wait_loadcnt.

typedef float v2f __attribute__((ext_vector_type(2)));
typedef float v4f __attribute__((ext_vector_type(4)));
typedef float v8f __attribute__((ext_vector_type(8)));

#define WAVES_PER_BLOCK 2
#define THREADS_PER_BLOCK (WAVES_PER_BLOCK * 32)
#define NUM_BLOCKS 4096
#define TILE_ROWS 16
#define ROW_STRIDE 132              // 128 + 4 dword pad: conflict-free b64 A-gather + b128 row reads
#define TILE_DWORDS (TILE_ROWS * ROW_STRIDE)

__global__ void WeightAndSum_zero_kernel(float* __restrict__ p, int n) {
    int i = blockIdx.x * blockDim.x + threadIdx.x;
    int stride = gridDim.x * blockDim.x;
    for (; i < n; i += stride) p[i] = 0.0f;
}

// ---- CDNA5 async-tensor path helpers (inline asm: portable across toolchains)
__device__ __forceinline__ void wait_asynccnt0()  { asm volatile("s_wait_asynccnt 0x0"  ::: "memory"); }
__device__ __forceinline__ void wait_asynccnt16() { asm volatile("s_wait_asynccnt 0x10" ::: "memory"); }
__device__ __forceinline__ void wait_dscnt0()     { asm volatile("s_wait_dscnt 0x0"     ::: "memory"); }

// Issue one 16-row tile (16 x 512B rows) as async global->LDS DMA.
// Per lane: 16B of row r -> padded LDS row (528B stride). ASYNCcnt += 16.
__device__ __forceinline__ void issue_tile_async(uint32_t lds0, int voff,
                                                 const float* __restrict__ feats) {
    #pragma unroll
    for (int r = 0; r < TILE_ROWS; ++r) {
        asm volatile("global_load_async_to_lds_b128 %0, %1, %2 th:TH_LOAD_NT"
                     :: "v"(lds0 + 528u * r),       // LDS byte addr (row stride 528)
                        "v"(voff + 512 * r),        // 32-bit global byte offset
                        "s"(feats)                  // 64-bit SGPR base
                     : "memory");
    }
}

__global__ __launch_bounds__(THREADS_PER_BLOCK)
void WeightAndSum_gate_segsum_kernel(const float* __restrict__ feats,
                                     const int* __restrict__ segids,
                                     const float* __restrict__ w,
                                     const float* __restrict__ bias,
                                     float* __restrict__ out,
                                     int N, int tilesPerWave, int numTiles,
                                     int fullTiles) {
    __shared__ float w_lds[128];
    __shared__ __align__(16) float tile_lds[WAVES_PER_BLOCK][2][TILE_DWORDS];

    const int tid = threadIdx.x;
    for (int i = tid; i < 128; i += THREADS_PER_BLOCK) w_lds[i] = w[i];
    __syncthreads();

    const int lane = tid & 31;
    const int wave = tid >> 5;
    const int gwave = blockIdx.x * WAVES_PER_BLOCK + wave;

    int t0 = gwave * tilesPerWave;
    int t1 = min(t0 + tilesPerWave, numTiles);
    if (t0 >= t1) return;               // wave-uniform exit; EXEC stays full for WMMA

    const float bval = bias[0];
    const int m = lane & 15;            // A-matrix row index held by this lane
    const int half = lane >> 4;         // 0: K={0,1}, 1: K={2,3} within a K=4 chunk

    float* buf0 = tile_lds[wave][0];
    float* buf1 = tile_lds[wave][1];
    // Wave-relative LDS byte addresses (low 32 bits of flat addr) for this lane
    const uint32_t lds_b[2] = {
        (uint32_t)(uintptr_t)(buf0 + lane * 4),
        (uint32_t)(uintptr_t)(buf1 + lane * 4)
    };
    const int lane_goff = lane * 16;

    v4f acc = {0.0f, 0.0f, 0.0f, 0.0f};
    int cur = segids[min(t0 * TILE_ROWS, N - 1)];

    // Prologue: prefetch first tile (if it is a full 16-row tile)
    if (t0 < fullTiles)
        issue_tile_async(lds_b[t0 & 1], t0 * (TILE_ROWS * 512) + lane_goff, feats);

    for (int t = t0; t < t1; ++t) {
        const int base = t * TILE_ROWS;
        float* my = (t & 1) ? buf1 : buf0;

        // Prefetch next tile into the other buffer (overlaps with compute below)
        const bool haveNext = (t + 1 < t1) && (t + 1 < fullTiles);
        if (haveNext) {
            wait_dscnt0();              // WAR: prior tile's ds reads vs incoming DMA
            issue_tile_async(lds_b[(t + 1) & 1],
                             (t + 1) * (TILE_ROWS * 512) + lane_goff, feats);
        }
        // Async loads retire in order: <=16 outstanding -> current tile landed
        if (haveNext) wait_asynccnt16(); else wait_asynccnt0();

        if (t >= fullTiles) {
            // partial tail tile: synchronous clamped staging
            for (int r = 0; r < TILE_ROWS; ++r) {
                int src = min(base + r, N - 1);
                const v4f* gp = (const v4f*)(feats + (size_t)src * 128 + lane * 4);
                v4f v = __builtin_nontemporal_load(gp);
                *(v4f*)(my + r * ROW_STRIDE + lane * 4) = v;
            }
        }

        // Pre-activation gates via WMMA: D = sum_kc A(16x4) x B(4x16) + C,
        // B[k][n] = w[4kc+k] for all n, C seeded with bias -> every column of
        // D holds gate_pre[row].
        v8f c;
        #pragma unroll
        for (int i = 0; i < 8; ++i) c[i] = bval;

        #pragma unroll
        for (int kc = 0; kc < 32; ++kc) {
            // A (f32 16x4): lanes 0-15 hold K=0,1; lanes 16-31 hold K=2,3
            v2f a = *(const v2f*)(my + m * ROW_STRIDE + kc * 4 + half * 2);
            // B (f32 4x16): VGPR0 = K=0 (lanes 0-15) / K=2 (lanes 16-31)
            v2f bm = *(const v2f*)(w_lds + kc * 4 + half * 2);
            c = __builtin_amdgcn_wmma_f32_16x16x4_f32(
                    false, a, false, bm, (short)0, c, false, false);
        }

        // sigmoid on the 8 D registers (rows j / j+8 per lane half)
        float g8[8];
        #pragma unroll
        for (int i = 0; i < 8; ++i) g8[i] = 1.0f / (1.0f + __expf(-c[i]));

        // segment ids for this tile: lane r (r<16) holds segids[base+r]
        int sid_l = segids[min(base + m, N - 1)];

        #pragma unroll
        for (int r = 0; r < TILE_ROWS; ++r) {
            int row = base + r;
            if (row >= N) break;                       // wave-uniform
            int sid = __shfl(sid_l, r, 32);
            // D layout: VGPR j = row j (lanes 0-15) / row j+8 (lanes 16-31)
            float g = __shfl(g8[r & 7], (r & 8) ? 16 : 0, 32);
            if (sid != cur) {                          // wave-uniform (sorted ids)
                float* o = out + (size_t)cur * 128 + lane * 4;
                unsafeAtomicAdd(o + 0, acc.x);
                unsafeAtomicAdd(o + 1, acc.y);
                unsafeAtomicAdd(o + 2, acc.z);
                unsafeAtomicAdd(o + 3, acc.w);
                acc = (v4f){0.0f, 0.0f, 0.0f, 0.0f};
                cur = sid;
            }
            v4f v = *(const v4f*)(my + r * ROW_STRIDE + lane * 4);
            acc += v * g;
        }
    }

    // final flush of the trailing segment
    float* o = out + (size_t)cur * 128 + lane * 4;
    unsafeAtomicAdd(o + 0, acc.x);
    unsafeAtomicAdd(o + 1, acc.y);
    unsafeAtomicAdd(o + 2, acc.z);
    unsafeAtomicAdd(o + 3, acc.w);
}

extern "C" void kernel_launch(void* const* d_in, const int* in_sizes, int n_in,
                              void* d_out, int out_size, void* d_ws, size_t ws_size,
                              hipStream_t stream) {
    const float* feats = (const float*)d_in[0];
    const int*   segid = (const int*)d_in[1];
    const float* w     = (const float*)d_in[2];
    const float* bias  = (const float*)d_in[3];
    float*       out   = (float*)d_out;

    const int N = in_sizes[1];            // number of rows (D = 128 fixed)

    // zero the (B,128) output before atomic accumulation
    WeightAndSum_zero_kernel<<<2048, 256, 0, stream>>>(out, out_size);

    const int numTiles  = (N + TILE_ROWS - 1) / TILE_ROWS;
    const int fullTiles = N / TILE_ROWS;
    const int totalWaves = NUM_BLOCKS * WAVES_PER_BLOCK;
    const int tilesPerWave = (numTiles + totalWaves - 1) / totalWaves;

    WeightAndSum_gate_segsum_kernel<<<NUM_BLOCKS, THREADS_PER_BLOCK, 0, stream>>>(
        feats, segid, w, bias, out, N, tilesPerWave, numTiles, fullTiles);
}